// gPool_71794673320391
// MI455X (gfx1250) — compile-verified
//
#include <hip/hip_runtime.h>

#define D_FEAT 256
#define NBINS 4096

typedef __attribute__((ext_vector_type(2))) float v2f;
typedef __attribute__((ext_vector_type(8))) float v8f;
typedef unsigned int u32;

// Order-preserving float -> uint key (descending score == descending key).
__device__ __forceinline__ u32 orderKey(float x) {
    u32 u = __float_as_uint(x);
    return (u & 0x80000000u) ? ~u : (u | 0x80000000u);
}

// ---------------------------------------------------------------------------
// Pass 1: logits = h @ W + b via V_WMMA_F32_16X16X4_F32.
// One wave per 16-row tile; 8 tiles per 256-thread block.
// B-tile replicates W[k..k+3] across all 16 N-columns, so every column of the
// f32 accumulator equals the row's logit.  C layout: lane 0 holds M=0..7 in
// acc[0..7], lane 16 holds M=8..15.
// The unrolled body is branch-free so the 8 independent global_load_b64 pairs
// can be issued ahead of the serial WMMA accumulation chain.
// ---------------------------------------------------------------------------
__global__ __launch_bounds__(256) void k_scores(const float* __restrict__ h,
                                                const float* __restrict__ W,
                                                const float* __restrict__ bptr,
                                                float* __restrict__ logits,
                                                int N) {
    const int lane = threadIdx.x & 31;
    const int wave = threadIdx.x >> 5;
    const int tile = blockIdx.x * 8 + wave;     // 16 rows per wave
    const int rbase = tile * 16;
    if (rbase >= N) return;                     // wave-uniform exit

    const int half = lane >> 4;                 // 0: K=0,1   1: K=2,3
    const int lrow = lane & 15;
    int row = rbase + lrow;
    if (row >= N) row = N - 1;                  // clamp: keep EXEC all-1s for WMMA
    const float* __restrict__ hrow = h + (size_t)row * D_FEAT;

    v8f acc = {};
#pragma unroll 8
    for (int k = 0; k < D_FEAT; k += 4) {
        // Constant-folded under unroll: one near-scope speculative prefetch per
        // 256B, no control flow (OOB prefetch past the last row is silently
        // dropped per the speculative-prefetch rules).
        if ((k & 63) == 0)
            __builtin_prefetch(hrow + k + 64, 0, 3);   // global_prefetch_b8
        const int kk = k + (half << 1);
        v2f a;  a.x = hrow[kk]; a.y = hrow[kk + 1];    // A: 16x4 f32 slab
        v2f bm; bm.x = W[kk];   bm.y = W[kk + 1];      // B: W replicated over N
        acc = __builtin_amdgcn_wmma_f32_16x16x4_f32(
            /*neg_a=*/false, a, /*neg_b=*/false, bm,
            /*c_mod=*/(short)0, acc, /*reuse_a=*/false, /*reuse_b=*/false);
    }

    const float bias = bptr[0];
    if (lrow == 0) {                            // lanes 0 and 16 own column N=0
        const int mofs = half * 8;
#pragma unroll
        for (int i = 0; i < 8; ++i) {
            const int r = rbase + mofs + i;
            if (r < N) logits[r] = acc[i] + bias;
        }
    }
}

// ---------------------------------------------------------------------------
// Top-k selection machinery (deterministic, atomic-free compaction).
// ---------------------------------------------------------------------------
__global__ void k_zero_hist(u32* hist) {
    hist[blockIdx.x * 256 + threadIdx.x] = 0;
}

__global__ void k_hist(const float* __restrict__ logits, u32* __restrict__ hist, int N) {
    int i = blockIdx.x * blockDim.x + threadIdx.x;
    if (i < N) atomicAdd(&hist[orderKey(logits[i]) >> 20], 1u);
}

// Serial descending scan over 4096 bins: find threshold bin T with
// count(bin > T) < K <= count(bin >= T).  meta = {T, above, need}.
__global__ void k_thresh(const u32* __restrict__ hist, u32* __restrict__ meta, int K) {
    if (threadIdx.x == 0) {
        u32 acc = 0, above = 0; int T = 0;
        for (int b = NBINS - 1; b >= 0; --b) {
            u32 c = hist[b];
            if (acc + c >= (u32)K) { T = b; above = acc; break; }
            acc += c;
            if (b == 0) { T = 0; above = acc; }
        }
        meta[0] = (u32)T;
        meta[1] = above;            // # strictly above threshold bin
        meta[2] = (u32)K - above;   // # needed from the tie bin
    }
}

// Per-node flags (1 = above bin, 2 = tie bin) + per-block (above, tie) counts.
__global__ __launch_bounds__(256) void k_flags_partials(const float* __restrict__ logits,
                                                        const u32* __restrict__ meta,
                                                        unsigned char* __restrict__ flags,
                                                        uint2* __restrict__ partials,
                                                        int N) {
    __shared__ u32 sA[8], sT[8];
    int i = blockIdx.x * 256 + threadIdx.x;
    u32 T = meta[0];
    bool fa = false, ft = false;
    if (i < N) {
        u32 bin = orderKey(logits[i]) >> 20;
        fa = bin > T;
        ft = bin == T;
        flags[i] = fa ? 1 : (ft ? 2 : 0);
    }
    unsigned long long mA = __ballot(fa);   // wave32: low 32 bits valid
    unsigned long long mT = __ballot(ft);
    int lane = threadIdx.x & 31, wid = threadIdx.x >> 5;
    if (lane == 0) { sA[wid] = __popcll(mA); sT[wid] = __popcll(mT); }
    __syncthreads();
    if (threadIdx.x == 0) {
        u32 a = 0, t = 0;
        for (int w = 0; w < 8; ++w) { a += sA[w]; t += sT[w]; }
        partials[blockIdx.x] = make_uint2(a, t);
    }
}

// Exclusive scan of block partials (391 entries -> serial is fine).
__global__ void k_scan_partials(uint2* partials, int nblocks) {
    if (threadIdx.x == 0) {
        u32 a = 0, t = 0;
        for (int i = 0; i < nblocks; ++i) {
            uint2 p = partials[i];
            partials[i] = make_uint2(a, t);
            a += p.x; t += p.y;
        }
    }
}

// Stable compaction: above-bin nodes at [0, A), tie-bin nodes at [A, K).
__global__ __launch_bounds__(256) void k_emit(const unsigned char* __restrict__ flags,
                                              const uint2* __restrict__ partials,
                                              const u32* __restrict__ meta,
                                              unsigned char* __restrict__ sel,
                                              int* __restrict__ slot_node,
                                              float* __restrict__ out_ids,
                                              int N, int K) {
    __shared__ u32 sA[8], sT[8];
    int i = blockIdx.x * 256 + threadIdx.x;
    unsigned char f = (i < N) ? flags[i] : (unsigned char)0;
    bool fa = (f == 1), ft = (f == 2);
    unsigned long long mA = __ballot(fa);
    unsigned long long mT = __ballot(ft);
    int lane = threadIdx.x & 31, wid = threadIdx.x >> 5;
    u32 lm = (1u << lane) - 1u;
    u32 pa = __popc((u32)mA & lm);
    u32 pt = __popc((u32)mT & lm);
    if (lane == 0) { sA[wid] = __popcll(mA); sT[wid] = __popcll(mT); }
    __syncthreads();
    if (threadIdx.x == 0) {
        u32 a = 0, t = 0;
        for (int w = 0; w < 8; ++w) {
            u32 x = sA[w], y = sT[w];
            sA[w] = a; sT[w] = t;
            a += x; t += y;
        }
    }
    __syncthreads();
    uint2 bo = partials[blockIdx.x];
    u32 aIdx = bo.x + sA[wid] + pa;
    u32 tIdx = bo.y + sT[wid] + pt;
    u32 A = meta[1], need = meta[2];
    long pos = -1;
    if (fa)                      pos = (long)aIdx;
    else if (ft && tIdx < need)  pos = (long)(A + tIdx);
    if (i < N) {
        unsigned char s = 0;
        if (pos >= 0 && pos < K) {
            s = 1;
            slot_node[pos] = i;
            out_ids[pos] = (float)i;
        }
        sel[i] = s;
    }
}

// Gather + gate: one block per selected row, 64 threads x float4 = 256 floats.
__global__ __launch_bounds__(64) void k_gather(const float* __restrict__ h,
                                               const float* __restrict__ logits,
                                               const int* __restrict__ slot_node,
                                               float* __restrict__ out_h) {
    int slot = blockIdx.x;
    int node = slot_node[slot];
    float lg = logits[node];
    float s = 1.0f / (1.0f + __expf(-lg));       // sigmoid applied post-selection
    const float4* __restrict__ src = (const float4*)(h + (size_t)node * D_FEAT);
    float4* __restrict__ dst = (float4*)(out_h + (size_t)slot * D_FEAT);
    float4 v = src[threadIdx.x];
    v.x *= s; v.y *= s; v.z *= s; v.w *= s;
    dst[threadIdx.x] = v;
}

// Edge mask: both endpoints selected.
__global__ void k_edges(const int* __restrict__ ei,
                        const unsigned char* __restrict__ sel,
                        float* __restrict__ out_mask, int E) {
    int e = blockIdx.x * blockDim.x + threadIdx.x;
    if (e < E) {
        int s = ei[e];
        int d = ei[E + e];
        out_mask[e] = (sel[s] && sel[d]) ? 1.0f : 0.0f;
    }
}

// ---------------------------------------------------------------------------
extern "C" void kernel_launch(void* const* d_in, const int* in_sizes, int n_in,
                              void* d_out, int out_size, void* d_ws, size_t ws_size,
                              hipStream_t stream) {
    const float* h = (const float*)d_in[0];
    const float* W = (const float*)d_in[1];
    const float* b = (const float*)d_in[2];
    const int*   ei = (const int*)d_in[3];

    const int N = in_sizes[0] / D_FEAT;          // 100000
    const int E = in_sizes[3] / 2;               // 3200000
    int K = (out_size - E) / (D_FEAT + 1);       // K*256 + K + E == out_size
    if (K <= 0 || K > N) K = 50000;

    // Workspace layout (256B-aligned slices).
    char* wsb = (char*)d_ws;
    auto alignup = [](size_t x) { return (x + 255) & ~(size_t)255; };
    size_t off = 0;
    float* logits = (float*)(wsb + off);         off = alignup(off + (size_t)N * 4);
    u32*   hist   = (u32*)(wsb + off);           off = alignup(off + (size_t)NBINS * 4);
    u32*   meta   = (u32*)(wsb + off);           off = alignup(off + 64);
    const int nsb = (N + 255) / 256;             // scan blocks
    uint2* partials = (uint2*)(wsb + off);       off = alignup(off + (size_t)nsb * 8);
    unsigned char* flags = (unsigned char*)(wsb + off); off = alignup(off + (size_t)N);
    unsigned char* sel   = (unsigned char*)(wsb + off); off = alignup(off + (size_t)N);
    int* slot_node = (int*)(wsb + off);          off = alignup(off + (size_t)K * 4);

    // Output layout: new_h [K*256] | node_ids [K] | edge_mask [E], all float32.
    float* out_h    = (float*)d_out;
    float* out_ids  = out_h + (size_t)K * D_FEAT;
    float* out_mask = out_ids + K;

    k_zero_hist<<<NBINS / 256, 256, 0, stream>>>(hist);

    const int ntiles = (N + 15) / 16;
    k_scores<<<(ntiles + 7) / 8, 256, 0, stream>>>(h, W, b, logits, N);

    k_hist<<<(N + 255) / 256, 256, 0, stream>>>(logits, hist, N);
    k_thresh<<<1, 32, 0, stream>>>(hist, meta, K);
    k_flags_partials<<<nsb, 256, 0, stream>>>(logits, meta, flags, partials, N);
    k_scan_partials<<<1, 32, 0, stream>>>(partials, nsb);
    k_emit<<<nsb, 256, 0, stream>>>(flags, partials, meta, sel, slot_node, out_ids, N, K);
    k_gather<<<K, 64, 0, stream>>>(h, logits, slot_node, out_h);
    k_edges<<<(E + 255) / 256, 256, 0, stream>>>(ei, sel, out_mask, E);
}